// Block_6717328851867
// MI455X (gfx1250) — compile-verified
//
#include <hip/hip_runtime.h>
#include <stdint.h>

#define AS1 __attribute__((address_space(1)))
#define AS3 __attribute__((address_space(3)))

typedef float v4f __attribute__((ext_vector_type(4)));

namespace {
constexpr unsigned kBlock = 256;             // 8 wave32 per workgroup
constexpr unsigned kItems = 8;               // timesteps per thread
constexpr unsigned kChunk = kBlock * kItems; // 2048 timesteps per tile
constexpr unsigned kWaves = kBlock / 32;     // 8
}

static_assert(kItems == 8, "float4 store paths assume 8 items/thread");

// gfx1250 async global->LDS copy (ASYNCcnt-tracked). Inline asm so it works on
// both ROCm 7.2 and the amdgpu-toolchain lane regardless of builtin arity.
__device__ __forceinline__ void async_g2l_b128(const float* g, float* l) {
  asm volatile("global_load_async_to_lds_b128 %0, %1, off"
               :
               : "v"((AS3 void*)l), "v"(g)
               : "memory");
}

__device__ __forceinline__ void wait_asynccnt0() {
  asm volatile("s_wait_asynccnt 0" ::: "memory");
}

// streaming (non-temporal) 128-bit store: outputs are write-once, keep them
// out of L2 so the 192MB cache stays available for input traffic
__device__ __forceinline__ void store_nt4(float* p, float a, float b, float c,
                                          float d) {
  v4f val = {a, b, c, d};
  __builtin_nontemporal_store(val, (v4f*)p);
}

__global__ void __launch_bounds__(kBlock)
snn_fused_scan(const float* __restrict__ current,   // [R, T]
               const float* __restrict__ beta_arr,  // [N]
               const float* __restrict__ v_init,    // [R]
               const float* __restrict__ vth_arr,   // [N]
               float* __restrict__ out_spk,         // [R, T]
               float* __restrict__ out_z,           // [R, T]
               float* __restrict__ out_mem,         // [R, T]
               int N, int T) {
  __shared__ float smem[kChunk] __attribute__((aligned(16)));
  __shared__ float waveQ[kWaves];
  __shared__ float waveS[kWaves];
  __shared__ float waveZ[kWaves];
  __shared__ float carry3[3];  // v_end, S_chunk, Z_chunk

  const unsigned tid  = threadIdx.x;
  const unsigned lane = tid & 31u;
  const unsigned wv   = tid >> 5;
  const unsigned t0   = tid * kItems;
  const int n = (int)blockIdx.x;            // channel (no integer division)
  const int r = (int)blockIdx.y * N + n;    // flat row in [0, B*N)

  const float beta = beta_arr[n];
  const float vth  = vth_arr[n];
  const float vini = v_init[r];

  const size_t rowoff = (size_t)r * (size_t)T;
  const float* row = current + rowoff;
  float* oS = out_spk + rowoff;
  float* oZ = out_z   + rowoff;
  float* oM = out_mem + rowoff;

  const float b2 = beta * beta;
  const float b4 = b2 * b2;
  const float b8 = b4 * b4;  // per-thread segment coefficient

  float vcar = 0.f;  // membrane state entering next chunk
  float Scar = 0.f;  // total spikes before chunk
  float Zcar = 0.f;  // total z (= cumsum(z1)) before chunk

  for (int base = 0; base < T; base += (int)kChunk) {
    const bool full = (base + (int)kChunk) <= T;

    // ---- stage one 2048-step tile of `current` into LDS ----
    if (full) {
      // two coalesced 128-bit async loads per lane: wave covers 512 B/round
      async_g2l_b128(row + base + (size_t)(tid * 4u), &smem[tid * 4u]);
      async_g2l_b128(row + base + (size_t)(kChunk / 2 + tid * 4u),
                     &smem[kChunk / 2 + tid * 4u]);
      wait_asynccnt0();
    } else {  // ragged tail (not hit for T=2048)
      for (unsigned k = tid; k < kChunk; k += kBlock)
        smem[k] = (base + (int)k < T) ? row[base + k] : 0.f;
    }
    __syncthreads();

    float c[kItems];
#pragma unroll
    for (unsigned j = 0; j < kItems; ++j) c[j] = smem[t0 + j];
    if (base == 0 && tid == 0) c[0] = fmaf(beta, vini, c[0]);  // inject v_init

    // ================= Phase A: membrane linear recurrence =================
    // local 8-step scan from zero state; endpoint = transform offset
    float v = 0.f;
#pragma unroll
    for (unsigned j = 0; j < kItems; ++j) v = fmaf(beta, v, c[j]);

    // inclusive lane scan of endpoints with uniform coefficient b8:
    // at distance d every combining lane's own segment has length exactly d,
    // so q = b8^d * q(lane-d) + q.  Also build b8^lane from the same powers.
    float q = v;
    float coef = b8;      // b8^d
    float powlane = 1.f;  // b8^lane
#pragma unroll
    for (unsigned d = 1; d < 32; d <<= 1) {
      const float up = __shfl_up(q, d, 32);
      if (lane & d) powlane *= coef;
      if (lane >= d) q = fmaf(coef, up, q);
      coef *= coef;
    }
    const float c32 = coef;  // b8^32 = per-wave coefficient

    if (lane == 31u) waveQ[wv] = q;
    __syncthreads();

    // wave-entry state (seeded with the chunk carry), <=7 iterations
    float Cw = vcar;
    for (unsigned w2 = 0; w2 < wv; ++w2) Cw = fmaf(c32, Cw, waveQ[w2]);
    float qex = __shfl_up(q, 1, 32);
    if (lane == 0u) qex = 0.f;
    const float seed = fmaf(powlane, Cw, qex);  // exact state entering thread

    // re-run the recurrence seeded with the global prefix; fuse spikes + the
    // thread-local double cumsum in the same pass
    float mv[kItems];
    float z1 = 0.f, z2 = 0.f;
    v = seed;
#pragma unroll
    for (unsigned j = 0; j < kItems; ++j) {
      v = fmaf(beta, v, c[j]);
      mv[j] = v;
      const float s = ((v - vth) > 0.f) ? 1.f : 0.f;
      z1 += s;
      z2 += z1;
      c[j] = z2;  // reuse registers: local z2 prefix
    }

    if (full) {
      store_nt4(oM + base + t0,     mv[0], mv[1], mv[2], mv[3]);
      store_nt4(oM + base + t0 + 4, mv[4], mv[5], mv[6], mv[7]);
    } else {
#pragma unroll
      for (unsigned j = 0; j < kItems; ++j)
        if (base + (int)(t0 + j) < T) oM[base + t0 + j] = mv[j];
    }

    // ============ Phase B: inter-thread scan of (S, Z) pairs ============
    // combine(prev, cur): Z = Zp + Zc + Sp*len(cur); S = Sp + Sc
    float S = z1, Z = z2;
#pragma unroll
    for (unsigned d = 1; d < 32; d <<= 1) {
      const float Sup = __shfl_up(S, d, 32);
      const float Zup = __shfl_up(Z, d, 32);
      if (lane >= d) {
        Z = Zup + Z + Sup * (float)(d * kItems);
        S = Sup + S;
      }
    }
    if (lane == 31u) { waveS[wv] = S; waveZ[wv] = Z; }
    __syncthreads();

    float Cs = 0.f, Cz = 0.f;  // exclusive prefix over earlier waves
    for (unsigned w2 = 0; w2 < wv; ++w2) {
      Cz += waveZ[w2] + Cs * (float)(32u * kItems);
      Cs += waveS[w2];
    }
    float Sex = __shfl_up(S, 1, 32);
    float Zex = __shfl_up(Z, 1, 32);
    if (lane == 0u) { Sex = 0.f; Zex = 0.f; }
    const float Spin = Cs + Sex;                               // chunk-internal
    const float Zpin = Cz + Zex + Cs * (float)(lane * kItems);
    const float Sx = Scar + Spin;                              // + chunk carry
    const float Zx = Zcar + Zpin + Scar * (float)t0;

    float zg[kItems], sg[kItems];
#pragma unroll
    for (unsigned j = 0; j < kItems; ++j) {
      const float zz = c[j] + Zx + Sx * (float)(j + 1);  // exact integers
      zg[j] = zz;
      sg[j] = (zz == 1.0f) ? 1.f : 0.f;
    }

    if (full) {
      store_nt4(oZ + base + t0,     zg[0], zg[1], zg[2], zg[3]);
      store_nt4(oZ + base + t0 + 4, zg[4], zg[5], zg[6], zg[7]);
      store_nt4(oS + base + t0,     sg[0], sg[1], sg[2], sg[3]);
      store_nt4(oS + base + t0 + 4, sg[4], sg[5], sg[6], sg[7]);
    } else {
#pragma unroll
      for (unsigned j = 0; j < kItems; ++j)
        if (base + (int)(t0 + j) < T) {
          oZ[base + t0 + j] = zg[j];
          oS[base + t0 + j] = sg[j];
        }
    }

    // ---- chunk carries for the next tile ----
    if (tid == kBlock - 1u) {
      carry3[0] = v;                                   // exact membrane state
      carry3[1] = Cs + S;                              // chunk spike total
      carry3[2] = Cz + Z + Cs * (float)(32u * kItems); // chunk z total
    }
    __syncthreads();
    vcar = carry3[0];
    const float Sch = carry3[1];
    const float Zch = carry3[2];
    Zcar = Zcar + Zch + Scar * (float)kChunk;
    Scar = Scar + Sch;
    // earlier syncs cover smem reuse; next iteration's first __syncthreads() protects carry reuse
  }
}

extern "C" void kernel_launch(void* const* d_in, const int* in_sizes, int n_in,
                              void* d_out, int out_size, void* d_ws, size_t ws_size,
                              hipStream_t stream) {
  (void)n_in; (void)d_ws; (void)ws_size; (void)out_size;
  const float* current = (const float*)d_in[0];  // [B,N,T]
  const float* beta    = (const float*)d_in[1];  // [N]
  const float* v_init  = (const float*)d_in[2];  // [B,N]
  const float* v_th    = (const float*)d_in[3];  // [N]

  const int N = in_sizes[1];            // 1024
  const int R = in_sizes[2];            // B*N = 16384 rows
  const int B = R / N;                  // 16
  const int T = in_sizes[0] / R;        // 2048

  float* out = (float*)d_out;
  float* oS = out;                          // spikes_out
  float* oZ = out + (size_t)R * (size_t)T;  // z
  float* oM = oZ  + (size_t)R * (size_t)T;  // membrane

  snn_fused_scan<<<dim3((unsigned)N, (unsigned)B), dim3(kBlock), 0, stream>>>(
      current, beta, v_init, v_th, oS, oZ, oM, N, T);
}